// Model_33217277067319
// MI455X (gfx1250) — compile-verified
//
#include <hip/hip_runtime.h>
#include <math.h>

#define NN 100000
#define EE 300000
#define DD 128

typedef float v2f __attribute__((ext_vector_type(2)));
typedef float v8f __attribute__((ext_vector_type(8)));

static __device__ __forceinline__ v8f v8zero() {
  v8f z;
#pragma unroll
  for (int r = 0; r < 8; ++r) z[r] = 0.0f;
  return z;
}

static __device__ __forceinline__ void atomAddF(float* p, float v) {
  unsafeAtomicAdd(p, v);   // global_atomic_add_f32
}

// gate code -> processing index: and=3 ->0, not=2 ->1, or=4 ->2, maj=1 ->3
static __device__ __forceinline__ int bucket_of(int lv, int code) {
  if (lv < 1 || lv > 6) return -1;
  int gi;
  switch (code) {
    case 3: gi = 0; break;
    case 2: gi = 1; break;
    case 4: gi = 2; break;
    case 1: gi = 3; break;
    default: return -1;
  }
  return (lv - 1) * 4 + gi;
}

// One 16(rows) x 16(cols n0..n0+15) tile of A(16xK, LDS) @ W(KxD, global),
// accumulated with V_WMMA_F32_16X16X4_F32 over K in steps of 4.
static __device__ __forceinline__ v8f wmma_tile(const float* __restrict__ As, int lda,
                                                const float* __restrict__ W, int ldw,
                                                int K, int n0, v8f acc) {
  const int l = threadIdx.x & 31;
  const int row = l & 15;            // A row for this lane
  const int koff = (l >> 4) << 1;    // 0 or 2
  const int col = n0 + row;          // B col for this lane
  for (int k = 0; k < K; k += 4) {
    v2f a, b;
    a.x = As[row * lda + k + koff];
    a.y = As[row * lda + k + koff + 1];
    b.x = W[(size_t)(k + koff) * ldw + col];
    b.y = W[(size_t)(k + koff + 1) * ldw + col];
    acc = __builtin_amdgcn_wmma_f32_16x16x4_f32(false, a, false, b, (short)0, acc,
                                                false, false);
  }
  return acc;
}

// ---------------- elementwise / setup kernels ----------------

__global__ void k_zero_f32(float* p, long n) {
  long i = (long)blockIdx.x * blockDim.x + threadIdx.x;
  if (i < n) p[i] = 0.0f;
}
__global__ void k_fill_f32(float* p, long n, float v) {
  long i = (long)blockIdx.x * blockDim.x + threadIdx.x;
  if (i < n) p[i] = v;
}
__global__ void k_zero_i32(int* p, int n) {
  int i = blockIdx.x * blockDim.x + threadIdx.x;
  if (i < n) p[i] = 0;
}

__global__ void k_deg(const int* __restrict__ dst, float* deg) {
  int e = blockIdx.x * blockDim.x + threadIdx.x;
  if (e < EE) atomAddF(&deg[dst[e]], 1.0f);
}

__global__ void k_coef(const int* __restrict__ src, const int* __restrict__ dst,
                       const int* __restrict__ flev, const float* __restrict__ deg,
                       float* coef) {
  int e = blockIdx.x * blockDim.x + threadIdx.x;
  if (e < EE) {
    int s = src[e], d = dst[e];
    if (flev[s] == 0) atomAddF(&coef[d], rsqrtf(deg[s]) * rsqrtf(deg[d]));
  }
}

// u = (hs_init@gcn_w)@mu_w ; v = gcn_b@mu_w + mu_b   (one block, 128 threads)
__global__ void k_guv(const float* __restrict__ hs_init, const float* __restrict__ gcn_w,
                      const float* __restrict__ gcn_b, const float* __restrict__ mu_w,
                      const float* __restrict__ mu_b, float* u, float* v) {
  __shared__ float sg[DD];
  int j = threadIdx.x;
  float s = 0.f;
  for (int k = 0; k < DD; ++k) s += hs_init[k] * gcn_w[k * DD + j];
  sg[j] = s;
  __syncthreads();
  float uu = 0.f, vv = 0.f;
  for (int k = 0; k < DD; ++k) {
    uu += sg[k] * mu_w[k * DD + j];
    vv += gcn_b[k] * mu_w[k * DD + j];
  }
  u[j] = uu;
  v[j] = vv + mu_b[j];
}

// hs[i] = (coef[i] + pi[i]/deg[i]) * u + v
__global__ void k_init_hs(const float* __restrict__ coef, const float* __restrict__ deg,
                          const int* __restrict__ flev, const float* __restrict__ u,
                          const float* __restrict__ v, float* hs) {
  long idx = (long)blockIdx.x * blockDim.x + threadIdx.x;
  if (idx >= (long)NN * 32) return;
  int n = (int)(idx >> 5);
  int c = (int)(idx & 31) << 2;
  float cf = coef[n] + (flev[n] == 0 ? 1.0f / deg[n] : 0.0f);
#pragma unroll
  for (int t = 0; t < 4; ++t) hs[(size_t)n * DD + c + t] = cf * u[c + t] + v[c + t];
}

// ---------------- bucket list construction ----------------

__global__ void k_count_edges(const int* __restrict__ dst, const int* __restrict__ flev,
                              const int* __restrict__ gate, int* ecnt) {
  int e = blockIdx.x * blockDim.x + threadIdx.x;
  if (e < EE) {
    int d = dst[e];
    int b = bucket_of(flev[d], gate[d]);
    if (b >= 0) atomicAdd(&ecnt[b], 1);
  }
}
__global__ void k_count_nodes(const int* __restrict__ flev, const int* __restrict__ gate,
                              int* ncnt) {
  int n = blockIdx.x * blockDim.x + threadIdx.x;
  if (n < NN) {
    int b = bucket_of(flev[n], gate[n]);
    if (b >= 0) atomicAdd(&ncnt[b], 1);
  }
}
__global__ void k_scan(const int* ecnt, int* eoff, int* ecur, const int* ncnt, int* noff,
                       int* ncur) {
  if (threadIdx.x == 0 && blockIdx.x == 0) {
    int ea = 0, na = 0;
    for (int b = 0; b < 24; ++b) {
      eoff[b] = ea; ecur[b] = ea; ea += ecnt[b];
      noff[b] = na; ncur[b] = na; na += ncnt[b];
    }
    eoff[24] = ea; noff[24] = na;
  }
}
__global__ void k_fill_edges(const int* __restrict__ dst, const int* __restrict__ flev,
                             const int* __restrict__ gate, int* ecur, int* elist) {
  int e = blockIdx.x * blockDim.x + threadIdx.x;
  if (e < EE) {
    int d = dst[e];
    int b = bucket_of(flev[d], gate[d]);
    if (b >= 0) elist[atomicAdd(&ecur[b], 1)] = e;
  }
}
__global__ void k_fill_nodes(const int* __restrict__ flev, const int* __restrict__ gate,
                             int* ncur, int* nlist) {
  int n = blockIdx.x * blockDim.x + threadIdx.x;
  if (n < NN) {
    int b = bucket_of(flev[n], gate[n]);
    if (b >= 0) nlist[atomicAdd(&ncur[b], 1)] = n;
  }
}

// zero / copy rows selected by node list
__global__ void k_zero_rows(float* buf, const int* __restrict__ nlist, const int* noffp,
                            const int* ncntp) {
  int off = *noffp;
  long total = (long)(*ncntp) * 32;
  long stride = (long)gridDim.x * blockDim.x;
  for (long i = (long)blockIdx.x * blockDim.x + threadIdx.x; i < total; i += stride) {
    int node = nlist[off + (int)(i >> 5)];
    int c = (int)(i & 31) << 2;
    *(float4*)&buf[(size_t)node * DD + c] = make_float4(0.f, 0.f, 0.f, 0.f);
  }
}
__global__ void k_copy_rows(float* dstbuf, const float* __restrict__ srcbuf,
                            const int* __restrict__ nlist, const int* noffp,
                            const int* ncntp) {
  int off = *noffp;
  long total = (long)(*ncntp) * 32;
  long stride = (long)gridDim.x * blockDim.x;
  for (long i = (long)blockIdx.x * blockDim.x + threadIdx.x; i < total; i += stride) {
    int node = nlist[off + (int)(i >> 5)];
    int c = (int)(i & 31) << 2;
    *(float4*)&dstbuf[(size_t)node * DD + c] = *(const float4*)&srcbuf[(size_t)node * DD + c];
  }
}

// ---------------- WMMA edge MLP + scatter: msg[dst] += (relu(X@W1+b1))@W2 + b2 -------
// KIN = 128: X = hs[src] ; KIN = 256: X = concat(hs[src], hf[src])
template <int KIN>
__global__ __launch_bounds__(32) void k_edge_mlp(
    const int* __restrict__ src, const int* __restrict__ dst, const int* __restrict__ elist,
    const int* eoffp, const int* ecntp, const float* __restrict__ hs,
    const float* __restrict__ hf, const float* __restrict__ W1, const float* __restrict__ b1,
    const float* __restrict__ W2, const float* __restrict__ b2, float* msg) {
  const int lda = KIN + 4;
  __shared__ float As[16 * (KIN + 4)];
  __shared__ float Hs[16 * 132];
  __shared__ int rowdst[16];

  const int l = threadIdx.x;
  const int row = l & 15, half = l >> 4;
  const int off = *eoffp, cnt = *ecntp;
  const int ntiles = (cnt + 15) >> 4;

  for (int tile = blockIdx.x; tile < ntiles; tile += gridDim.x) {
    int eidx = tile * 16 + row;
    int sn = -1, dn = -1;
    if (eidx < cnt) {
      int e = elist[off + eidx];
      sn = src[e];
      dn = dst[e];
    }
    if (half == 0) rowdst[row] = dn;
    // gather X row into LDS (each half-wave covers KIN/2 columns)
    for (int c = 0; c < KIN / 2; c += 4) {
      int col = half * (KIN / 2) + c;
      float4 val = make_float4(0.f, 0.f, 0.f, 0.f);
      if (sn >= 0) {
        if (KIN == 128)
          val = *(const float4*)&hs[(size_t)sn * DD + col];
        else
          val = (half == 0) ? *(const float4*)&hs[(size_t)sn * DD + c]
                            : *(const float4*)&hf[(size_t)sn * DD + c];
      }
      *(float4*)&As[row * lda + col] = val;
    }
    __syncthreads();

    const int mcol16 = l & 15, hi8 = (l >> 4) * 8;
    // layer 1: H = relu(X @ W1 + b1)
    for (int n0 = 0; n0 < DD; n0 += 16) {
      v8f acc = v8zero();
      acc = wmma_tile(As, lda, W1, DD, KIN, n0, acc);
      int mcol = n0 + mcol16;
      float bb = b1[mcol];
#pragma unroll
      for (int r = 0; r < 8; ++r) Hs[(hi8 + r) * 132 + mcol] = fmaxf(acc[r] + bb, 0.f);
    }
    __syncthreads();
    // layer 2: scatter (H @ W2 + b2) into msg[dst]
    for (int n0 = 0; n0 < DD; n0 += 16) {
      v8f acc = v8zero();
      acc = wmma_tile(Hs, 132, W2, DD, DD, n0, acc);
      int mcol = n0 + mcol16;
      float bb = b2[mcol];
#pragma unroll
      for (int r = 0; r < 8; ++r) {
        int dnode = rowdst[hi8 + r];
        if (dnode >= 0) atomAddF(&msg[(size_t)dnode * DD + mcol], acc[r] + bb);
      }
    }
    __syncthreads();
  }
}

// hs[n] = act(concat(hs[n], msg[n]) @ UW + ub) for nodes in list
__global__ __launch_bounds__(32) void k_upd(const int* __restrict__ nlist, const int* noffp,
                                            const int* ncntp, float* hs,
                                            const float* __restrict__ msg,
                                            const float* __restrict__ UW,
                                            const float* __restrict__ ub, int act_tanh) {
  __shared__ float As[16 * 260];
  __shared__ int rown[16];
  const int l = threadIdx.x;
  const int row = l & 15, half = l >> 4;
  const int off = *noffp, cnt = *ncntp;
  const int ntiles = (cnt + 15) >> 4;

  for (int tile = blockIdx.x; tile < ntiles; tile += gridDim.x) {
    int nidx = tile * 16 + row;
    int node = (nidx < cnt) ? nlist[off + nidx] : -1;
    if (half == 0) rown[row] = node;
    const float* sb = (half == 0) ? hs : msg;
    int cb = half * DD;
    for (int c = 0; c < DD; c += 4) {
      float4 val = make_float4(0.f, 0.f, 0.f, 0.f);
      if (node >= 0) val = *(const float4*)&sb[(size_t)node * DD + c];
      *(float4*)&As[row * 260 + cb + c] = val;
    }
    __syncthreads();
    const int mcol16 = l & 15, hi8 = (l >> 4) * 8;
    for (int n0 = 0; n0 < DD; n0 += 16) {
      v8f acc = v8zero();
      acc = wmma_tile(As, 260, UW, DD, 256, n0, acc);
      int mcol = n0 + mcol16;
      float bb = ub[mcol];
#pragma unroll
      for (int r = 0; r < 8; ++r) {
        int nd = rown[hi8 + r];
        if (nd >= 0) {
          float vv = acc[r] + bb;
          vv = act_tanh ? tanhf(vv) : fmaxf(vv, 0.f);
          hs[(size_t)nd * DD + mcol] = vv;
        }
      }
    }
    __syncthreads();
  }
}

// out = hs @ W  (dense over all N rows)
__global__ __launch_bounds__(32) void k_dense(const float* __restrict__ hs,
                                              const float* __restrict__ W, float* out) {
  __shared__ float As[16 * 132];
  const int l = threadIdx.x;
  const int row = l & 15, half = l >> 4;
  const int ntiles = (NN + 15) >> 4;
  for (int tile = blockIdx.x; tile < ntiles; tile += gridDim.x) {
    int node = tile * 16 + row;
    int cb = half * 64;
    for (int c = 0; c < 64; c += 4) {
      float4 val = make_float4(0.f, 0.f, 0.f, 0.f);
      if (node < NN) val = *(const float4*)&hs[(size_t)node * DD + cb + c];
      *(float4*)&As[row * 132 + cb + c] = val;
    }
    __syncthreads();
    const int mcol16 = l & 15, hi8 = (l >> 4) * 8;
    for (int n0 = 0; n0 < DD; n0 += 16) {
      v8f acc = v8zero();
      acc = wmma_tile(As, 132, W, DD, DD, n0, acc);
#pragma unroll
      for (int r = 0; r < 8; ++r) {
        int nd = tile * 16 + hi8 + r;
        if (nd < NN) out[(size_t)nd * DD + n0 + mcol16] = acc[r];
      }
    }
    __syncthreads();
  }
}

// hf[dst] += sigmoid(zs[src] * zt[dst]) elementwise
__global__ void k_decoder(const int* __restrict__ src, const int* __restrict__ dst,
                          const float* __restrict__ zs, const float* __restrict__ zt,
                          float* hf) {
  long idx = (long)blockIdx.x * blockDim.x + threadIdx.x;
  if (idx >= (long)EE * 32) return;
  int e = (int)(idx >> 5);
  int c = (int)(idx & 31) << 2;
  int s = src[e], d = dst[e];
#pragma unroll
  for (int t = 0; t < 4; ++t) {
    float x = zs[(size_t)s * DD + c + t] * zt[(size_t)d * DD + c + t];
    float sg = 1.0f / (1.0f + expf(-x));
    atomAddF(&hf[(size_t)d * DD + c + t], sg);
  }
}

static inline int cdiv(long a, int b) { return (int)((a + b - 1) / b); }

extern "C" void kernel_launch(void* const* d_in, const int* in_sizes, int n_in, void* d_out,
                              int out_size, void* d_ws, size_t ws_size, hipStream_t stream) {
  const int* ei = (const int*)d_in[0];
  const int* src = ei;
  const int* dst = ei + EE;
  const int* gate = (const int*)d_in[1];
  const int* flev = (const int*)d_in[2];
  const float* hs_init = (const float*)d_in[3];
  const float* gcn_w = (const float*)d_in[4];
  const float* gcn_b = (const float*)d_in[5];
  const float* mu_w = (const float*)d_in[6];
  const float* mu_b = (const float*)d_in[7];
  // d_in[8]/[9] = ls_w/ls_b (unused in eval mode)
  const float* dec_ws = (const float*)d_in[50];
  const float* dec_wt = (const float*)d_in[51];

  float* hs = (float*)d_out;                 // final output 0
  float* hf = hs + (size_t)NN * DD;          // final output 1

  float* buf0 = (float*)d_ws;                // msg / zs  (N*D)
  float* buf1 = buf0 + (size_t)NN * DD;      // zt        (N*D)
  float* deg = buf1 + (size_t)NN * DD;       // N
  float* coef = deg + NN;                    // N
  float* uvec = coef + NN;                   // D
  float* vvec = uvec + DD;                   // D
  int* ibase = (int*)(vvec + DD);
  int* ecnt = ibase;            // 24
  int* eoff = ecnt + 24;        // 25
  int* ecur = eoff + 25;        // 24
  int* ncnt = ecur + 24;        // 24
  int* noff = ncnt + 24;        // 25
  int* ncur = noff + 25;        // 24
  int* elist = ncur + 24;       // E
  int* nlist = elist + EE;      // N

  // ---- init hs (rank-1 GCN+mu collapse), zero hf ----
  k_fill_f32<<<cdiv(NN, 256), 256, 0, stream>>>(deg, NN, 1.0f);
  k_zero_f32<<<cdiv(NN, 256), 256, 0, stream>>>(coef, NN);
  k_zero_i32<<<1, 256, 0, stream>>>(ibase, 146);
  k_deg<<<cdiv(EE, 256), 256, 0, stream>>>(dst, deg);
  k_coef<<<cdiv(EE, 256), 256, 0, stream>>>(src, dst, flev, deg, coef);
  k_guv<<<1, 128, 0, stream>>>(hs_init, gcn_w, gcn_b, mu_w, mu_b, uvec, vvec);
  k_init_hs<<<cdiv((long)NN * 32, 256), 256, 0, stream>>>(coef, deg, flev, uvec, vvec, hs);
  k_zero_f32<<<cdiv((long)NN * DD, 256), 256, 0, stream>>>(hf, (long)NN * DD);

  // ---- build per-(level,gate) edge/node lists ----
  k_count_edges<<<cdiv(EE, 256), 256, 0, stream>>>(dst, flev, gate, ecnt);
  k_count_nodes<<<cdiv(NN, 256), 256, 0, stream>>>(flev, gate, ncnt);
  k_scan<<<1, 1, 0, stream>>>(ecnt, eoff, ecur, ncnt, noff, ncur);
  k_fill_edges<<<cdiv(EE, 256), 256, 0, stream>>>(dst, flev, gate, ecur, elist);
  k_fill_nodes<<<cdiv(NN, 256), 256, 0, stream>>>(flev, gate, ncur, nlist);

  // ---- 6 levels x 4 gates, sequential (order matters) ----
  for (int lv = 1; lv <= 6; ++lv) {
    for (int gidx = 0; gidx < 4; ++gidx) {
      int c = (lv - 1) * 4 + gidx;
      int pb = 10 + 10 * gidx;
      const float* sw1 = (const float*)d_in[pb + 0];
      const float* sb1 = (const float*)d_in[pb + 1];
      const float* sw2 = (const float*)d_in[pb + 2];
      const float* sb2 = (const float*)d_in[pb + 3];
      const float* fw1 = (const float*)d_in[pb + 4];
      const float* fb1 = (const float*)d_in[pb + 5];
      const float* fw2 = (const float*)d_in[pb + 6];
      const float* fb2 = (const float*)d_in[pb + 7];
      const float* uw = (const float*)d_in[pb + 8];
      const float* ub = (const float*)d_in[pb + 9];
      int act_tanh = (gidx == 1) ? 1 : 0;  // "not" gate uses tanh

      k_zero_rows<<<256, 256, 0, stream>>>(buf0, nlist, noff + c, ncnt + c);
      k_edge_mlp<128><<<2048, 32, 0, stream>>>(src, dst, elist, eoff + c, ecnt + c, hs,
                                               (const float*)nullptr, sw1, sb1, sw2, sb2, buf0);
      k_upd<<<1024, 32, 0, stream>>>(nlist, noff + c, ncnt + c, hs, buf0, uw, ub, act_tanh);
      k_zero_rows<<<256, 256, 0, stream>>>(buf0, nlist, noff + c, ncnt + c);
      k_edge_mlp<256><<<2048, 32, 0, stream>>>(src, dst, elist, eoff + c, ecnt + c, hs, hf,
                                               fw1, fb1, fw2, fb2, buf0);
      k_copy_rows<<<256, 256, 0, stream>>>(hf, buf0, nlist, noff + c, ncnt + c);
    }
  }

  // ---- decoder: zs = hs@dec_ws, zt = hs@dec_wt, hf = segsum(sigmoid(zs[s]*zt[d])) ----
  int dtiles = (NN + 15) / 16;
  k_dense<<<dtiles, 32, 0, stream>>>(hs, dec_ws, buf0);
  k_dense<<<dtiles, 32, 0, stream>>>(hs, dec_wt, buf1);
  k_zero_f32<<<cdiv((long)NN * DD, 256), 256, 0, stream>>>(hf, (long)NN * DD);
  k_decoder<<<cdiv((long)EE * 32, 256), 256, 0, stream>>>(src, dst, buf0, buf1, hf);
}